// GATNet_59528246722754
// MI455X (gfx1250) — compile-verified
//
#include <hip/hip_runtime.h>
#include <hip/hip_bf16.h>

#define NNODE 50000
#define NEDGE 400000
#define KFUSE 288   // 2*OUT + EIN
#define KAGG  320   // 2*OUT + 2*EIN
#define BPITCH 136  // LDS pitch (floats) for a 32x128 weight chunk (bank-conflict free)
#define BCHUNK (32 * BPITCH)   // one B buffer, in floats

typedef __attribute__((ext_vector_type(2))) float v2f;
typedef __attribute__((ext_vector_type(8))) float v8f;

// ---- order-preserving float<->int encoding for integer atomicMax segment-max ----
__device__ __forceinline__ int enc_ord(float f) {
  int i = __float_as_int(f);
  return i >= 0 ? i : (i ^ 0x7FFFFFFF);
}
__device__ __forceinline__ float dec_ord(int i) {
  return __int_as_float(i >= 0 ? i : (i ^ 0x7FFFFFFF));
}
#define ENC_NEG_INF ((int)0x807FFFFF)  // enc_ord(-inf)

// ---------------------------------------------------------------------------
// Async DMA of one 32x128 fp32 weight chunk into an LDS buffer using the
// CDNA5 async-to-LDS path (ASYNCcnt, no VGPR round trip).  256 threads x
// 4 issues x 16B = 16 KB.  Rows are 128 dwords so a B128 never crosses a row.
// ---------------------------------------------------------------------------
__device__ __forceinline__ void async_load_chunk(const float* __restrict__ Wg, int k0,
                                                 float* __restrict__ Bbuf, int tid) {
#pragma unroll
  for (int i = 0; i < 4; ++i) {
    const int d  = i * 1024 + tid * 4;   // dword index within the 32x128 chunk
    const int kk = d >> 7;
    const int n  = d & 127;
    const unsigned lds = (unsigned)(size_t)(void*)&Bbuf[kk * BPITCH + n];
    const float* g = Wg + (size_t)(k0 + kk) * 128 + n;
    asm volatile("global_load_async_to_lds_b128 %0, %1, off"
                 :: "v"(lds), "v"(g) : "memory");
  }
}
__device__ __forceinline__ void wait_async0() {
  asm volatile("s_wait_asynccnt 0x0" ::: "memory");
}

// ---------------------------------------------------------------------------
// Wave-level fp32 WMMA GEMM over an LDS A-tile (16 x K, pitch lda) and a
// weight matrix W (K x 128) double-buffer-streamed into LDS via async DMA.
// 8 waves per 256-thread block; wave w owns output columns [16w, 16w+16).
// Per ISA 7.12.2:  A 16x4 f32: lanes0-15 K={0,1}, lanes16-31 K={2,3};
//                  B 4x16 f32 mirrored;  C: VGPR i -> M=i (lanes0-15) / M=8+i.
// ---------------------------------------------------------------------------
__device__ __forceinline__ v8f lds_gemm(const float* __restrict__ As, int lda,
                                        const float* __restrict__ Wg, int K,
                                        float* __restrict__ Bs /* 2*BCHUNK */, int tid) {
  v8f acc = {0.f, 0.f, 0.f, 0.f, 0.f, 0.f, 0.f, 0.f};
  const int wave = tid >> 5;
  const int lane = tid & 31;
  const int half = lane >> 4;
  const int mr   = lane & 15;
  const int n0   = wave << 4;

  async_load_chunk(Wg, 0, Bs, tid);
  wait_async0();
  __syncthreads();

  int buf = 0;
  for (int k0 = 0; k0 < K; k0 += 32) {
    const float* Bcur = Bs + buf * BCHUNK;
    if (k0 + 32 < K)                      // overlap next chunk's DMA with WMMA
      async_load_chunk(Wg, k0 + 32, Bs + (buf ^ 1) * BCHUNK, tid);
#pragma unroll
    for (int kk = 0; kk < 32; kk += 4) {
      v2f a, b;
      const int ka = k0 + kk + 2 * half;
      a.x = As[mr * lda + ka];
      a.y = As[mr * lda + ka + 1];
      const int kb = kk + 2 * half;
      b.x = Bcur[kb * BPITCH + n0 + mr];
      b.y = Bcur[(kb + 1) * BPITCH + n0 + mr];
      acc = __builtin_amdgcn_wmma_f32_16x16x4_f32(false, a, false, b, (short)0,
                                                  acc, false, false);
    }
    wait_async0();
    __syncthreads();
    buf ^= 1;
  }
  return acc;
}

// ---------------------------------------------------------------------------
__global__ void k_init(float* __restrict__ den, float* __restrict__ deg,
                       int* __restrict__ mbuf, float* __restrict__ hagg) {
  long long i = (long long)blockIdx.x * 256 + threadIdx.x;
  if (i < NNODE) { den[i] = 0.f; deg[i] = 0.f; mbuf[i] = ENC_NEG_INF; }
  if (i < (long long)NNODE * 160) hagg[i] = 0.f;
}

// hn = h @ W_embed_node   (50000x128 @ 128x128)
__global__ void k_node_embed(const float* __restrict__ h, const float* __restrict__ W,
                             float* __restrict__ hn) {
  __shared__ float As[16 * 132];
  __shared__ float Bs[2 * BCHUNK];
  const int tid = threadIdx.x;
  const int r0 = blockIdx.x * 16;
  for (int idx = tid; idx < 16 * 128; idx += 256) {
    int r = idx >> 7, c = idx & 127;
    As[r * 132 + c] = h[(size_t)(r0 + r) * 128 + c];
  }
  __syncthreads();
  v8f acc = lds_gemm(As, 132, W, 128, Bs, tid);
  const int lane = tid & 31, wave = tid >> 5;
  const int half = lane >> 4, mr = lane & 15;
  const int col = (wave << 4) + mr;
#pragma unroll
  for (int i = 0; i < 8; ++i)
    hn[(size_t)(r0 + i + 8 * half) * 128 + col] = acc[i];
}

// ow = w @ W_embed_edge ; s = lrelu(ow)@W_edge_sf ; w1 = s*ow
__global__ void k_edge_embed(const float* __restrict__ w, const float* __restrict__ We,
                             const float* __restrict__ Wsf,
                             float* __restrict__ ow, float* __restrict__ w1) {
  __shared__ float sW[32 * 32];
  __shared__ float sSF[32];
  const int tid = threadIdx.x;
  for (int i = tid; i < 1024; i += 256) sW[i] = We[i];
  if (tid < 32) sSF[tid] = Wsf[tid];
  __syncthreads();
  const int e = blockIdx.x * 256 + tid;
  if (e >= NEDGE) return;
  float win[32];
#pragma unroll
  for (int i = 0; i < 32; i += 4) {
    const float4 v = *(const float4*)(w + (size_t)e * 32 + i);
    win[i] = v.x; win[i + 1] = v.y; win[i + 2] = v.z; win[i + 3] = v.w;
  }
  float o[32];
  float s = 0.f;
#pragma unroll 4
  for (int j = 0; j < 32; ++j) {
    float acc = 0.f;
#pragma unroll
    for (int i = 0; i < 32; ++i) acc = fmaf(win[i], sW[i * 32 + j], acc);
    o[j] = acc;
    const float lr = acc > 0.f ? acc : 0.1f * acc;
    s = fmaf(lr, sSF[j], s);
  }
#pragma unroll
  for (int j = 0; j < 32; j += 4) {
    *(float4*)(ow + (size_t)e * 32 + j) = make_float4(o[j], o[j+1], o[j+2], o[j+3]);
    *(float4*)(w1 + (size_t)e * 32 + j) = make_float4(s*o[j], s*o[j+1], s*o[j+2], s*o[j+3]);
  }
}

// sat build + z2 GEMM + lrelu + attention dot + segment-max + degree + w2
__global__ void k_inter_fuse(const float* __restrict__ hn, const float* __restrict__ w1,
                             const int* __restrict__ src, const int* __restrict__ dst,
                             const float* __restrict__ Wf, const float* __restrict__ bf,
                             const float* __restrict__ Wattn,
                             float* __restrict__ sat, float* __restrict__ logits,
                             float* __restrict__ w2, int* __restrict__ mbuf,
                             float* __restrict__ deg) {
  __shared__ float As[16 * 292];
  __shared__ float Bs[2 * BCHUNK];
  __shared__ float INW[16 * 128];
  const int tid = threadIdx.x;
  const int e0 = blockIdx.x * 16;
  for (int idx = tid; idx < 16 * KFUSE; idx += 256) {
    const int r = idx / KFUSE, c = idx - r * KFUSE;
    const int e = e0 + r;
    float v;
    if (c < 128)      v = hn[(size_t)src[e] * 128 + c];
    else if (c < 256) v = hn[(size_t)dst[e] * 128 + (c - 128)];
    else              v = w1[(size_t)e * 32 + (c - 256)];
    As[r * 292 + c] = v;
    sat[(size_t)e * KFUSE + c] = v;   // third output, materialized here
  }
  __syncthreads();
  v8f acc = lds_gemm(As, 292, Wf, KFUSE, Bs, tid);
  const int lane = tid & 31, wave = tid >> 5;
  const int half = lane >> 4, mr = lane & 15;
  const int col = (wave << 4) + mr;
#pragma unroll
  for (int i = 0; i < 8; ++i) {
    const float z = acc[i] + bf[col];
    INW[(i + 8 * half) * 128 + col] = z > 0.f ? z : 0.1f * z;
  }
  __syncthreads();
  if (tid < 16) {
    const int e = e0 + tid;
    float att = 0.f;
#pragma unroll 8
    for (int c = 0; c < 128; ++c) att = fmaf(INW[tid * 128 + c], Wattn[c], att);
    logits[e] = att;
    atomicMax(&mbuf[dst[e]], enc_ord(att));
    atomicAdd(&deg[dst[e]], 1.0f);
    for (int c = 0; c < 32; ++c) w2[(size_t)e * 32 + c] = att * w1[(size_t)e * 32 + c];
  }
}

__global__ void k_softmax_den(const float* __restrict__ logits, const int* __restrict__ dst,
                              const int* __restrict__ mbuf, float* __restrict__ ex,
                              float* __restrict__ den) {
  const int e = blockIdx.x * 256 + threadIdx.x;
  if (e >= NEDGE) return;
  const float m = dec_ord(mbuf[dst[e]]);
  const float v = __expf(logits[e] - m);
  ex[e] = v;
  atomicAdd(&den[dst[e]], v);
}

// hagg[dst] += alpha * [hs(128), w2(32)].  dst is sorted, so run-length
// accumulate across this block's 16 edges and only flush an atomic when the
// destination changes (~8x fewer global atomics at mean degree 8).
__global__ void k_scatter(const float* __restrict__ sat, const float* __restrict__ w2,
                          const float* __restrict__ ex, const float* __restrict__ den,
                          const int* __restrict__ dst, float* __restrict__ hagg) {
  const int e0 = blockIdx.x * 16;
  const int c  = threadIdx.x;          // 160 threads, one column each
  int   curd = dst[e0];
  float accv = 0.f;
  for (int r = 0; r < 16; ++r) {
    const int e = e0 + r;
    const int d = dst[e];
    if (d != curd) {
      atomicAdd(&hagg[(size_t)curd * 160 + c], accv);
      accv = 0.f;
      curd = d;
    }
    const float alpha = ex[e] / den[d];
    const float v = (c < 128) ? sat[(size_t)e * KFUSE + c]   // hs re-read from sat
                              : w2[(size_t)e * 32 + (c - 128)];
    accv = fmaf(alpha, v, accv);
  }
  atomicAdd(&hagg[(size_t)curd * 160 + c], accv);
}

// hnew = where(deg>0, [hagg,hn]@Wc + bc, hn)
__global__ void k_concentrate(const float* __restrict__ hagg, const float* __restrict__ hn,
                              const float* __restrict__ Wc, const float* __restrict__ bc,
                              const float* __restrict__ deg, float* __restrict__ hnew) {
  __shared__ float As[16 * 292];
  __shared__ float Bs[2 * BCHUNK];
  const int tid = threadIdx.x;
  const int n0b = blockIdx.x * 16;
  for (int idx = tid; idx < 16 * KFUSE; idx += 256) {
    const int r = idx / KFUSE, c = idx - r * KFUSE;
    const int n = n0b + r;
    As[r * 292 + c] = (c < 160) ? hagg[(size_t)n * 160 + c]
                                : hn[(size_t)n * 128 + (c - 160)];
  }
  __syncthreads();
  v8f acc = lds_gemm(As, 292, Wc, KFUSE, Bs, tid);
  const int lane = tid & 31, wave = tid >> 5;
  const int half = lane >> 4, mr = lane & 15;
  const int col = (wave << 4) + mr;
#pragma unroll
  for (int i = 0; i < 8; ++i) {
    const int n = n0b + i + 8 * half;
    float z = acc[i] + bc[col];
    if (!(deg[n] > 0.f)) z = hn[(size_t)n * 128 + col];
    hnew[(size_t)n * 128 + col] = z;
  }
}

// wout = [hnew[src], hnew[dst], bn(w2), ow] @ W_aggre
__global__ void k_aggre(const float* __restrict__ hnew, const int* __restrict__ src,
                        const int* __restrict__ dst,
                        const float* __restrict__ w2, const float* __restrict__ ow,
                        const float* __restrict__ gamma, const float* __restrict__ beta,
                        const float* __restrict__ mean, const float* __restrict__ var,
                        const float* __restrict__ Wa, float* __restrict__ wout) {
  __shared__ float As[16 * 324];
  __shared__ float Bs[2 * BCHUNK];
  const int tid = threadIdx.x;
  const int e0 = blockIdx.x * 16;
  for (int idx = tid; idx < 16 * KAGG; idx += 256) {
    const int r = idx / KAGG, c = idx - r * KAGG;
    const int e = e0 + r;
    float v;
    if (c < 128)      v = hnew[(size_t)src[e] * 128 + c];
    else if (c < 256) v = hnew[(size_t)dst[e] * 128 + (c - 128)];
    else if (c < 288) {
      const int j = c - 256;
      v = (w2[(size_t)e * 32 + j] - mean[j]) * rsqrtf(var[j] + 1e-5f) * gamma[j] + beta[j];
    } else            v = ow[(size_t)e * 32 + (c - 288)];
    As[r * 324 + c] = v;
  }
  __syncthreads();
  v8f acc = lds_gemm(As, 324, Wa, KAGG, Bs, tid);
  const int lane = tid & 31, wave = tid >> 5;
  const int half = lane >> 4, mr = lane & 15;
  const int col = (wave << 4) + mr;
#pragma unroll
  for (int i = 0; i < 8; ++i)
    wout[(size_t)(e0 + i + 8 * half) * 128 + col] = acc[i];
}

// ---------------------------------------------------------------------------
extern "C" void kernel_launch(void* const* d_in, const int* in_sizes, int n_in,
                              void* d_out, int out_size, void* d_ws, size_t ws_size,
                              hipStream_t stream) {
  const float* h            = (const float*)d_in[0];
  const float* w            = (const float*)d_in[1];
  const int*   src          = (const int*)d_in[2];
  const int*   dst          = (const int*)d_in[3];
  const float* W_embed_node = (const float*)d_in[4];
  const float* W_attn       = (const float*)d_in[5];
  const float* W_inter_fuse = (const float*)d_in[6];
  const float* b_inter_fuse = (const float*)d_in[7];
  const float* W_embed_edge = (const float*)d_in[8];
  const float* W_edge_sf    = (const float*)d_in[9];
  const float* W_concentrate= (const float*)d_in[10];
  const float* b_concentrate= (const float*)d_in[11];
  const float* bn_gamma     = (const float*)d_in[12];
  const float* bn_beta      = (const float*)d_in[13];
  const float* bn_mean      = (const float*)d_in[14];
  const float* bn_var       = (const float*)d_in[15];
  const float* W_aggre      = (const float*)d_in[16];

  // outputs: (hnew 50000x128, wout 400000x128, sat 400000x288) flat
  float* out      = (float*)d_out;
  float* hnew_out = out;
  float* wout_out = out + (size_t)NNODE * 128;
  float* sat_out  = wout_out + (size_t)NEDGE * 128;

  // workspace layout (floats)
  float* ws    = (float*)d_ws;
  float* hn    = ws;                           // 50000*128
  float* ow    = hn    + (size_t)NNODE * 128;  // 400000*32
  float* w1    = ow    + (size_t)NEDGE * 32;   // 400000*32
  float* w2    = w1    + (size_t)NEDGE * 32;   // 400000*32
  float* logit = w2    + (size_t)NEDGE * 32;   // 400000
  float* ex    = logit + NEDGE;                // 400000
  float* den   = ex    + NEDGE;                // 50000
  float* deg   = den   + NNODE;                // 50000
  int*   mbuf  = (int*)(deg + NNODE);          // 50000
  float* hagg  = (float*)(mbuf + NNODE);       // 50000*160

  k_init<<<(NNODE * 160 + 255) / 256, 256, 0, stream>>>(den, deg, mbuf, hagg);
  k_node_embed<<<NNODE / 16, 256, 0, stream>>>(h, W_embed_node, hn);
  k_edge_embed<<<(NEDGE + 255) / 256, 256, 0, stream>>>(w, W_embed_edge, W_edge_sf, ow, w1);
  k_inter_fuse<<<NEDGE / 16, 256, 0, stream>>>(hn, w1, src, dst, W_inter_fuse,
                                               b_inter_fuse, W_attn, sat_out, logit,
                                               w2, mbuf, deg);
  k_softmax_den<<<(NEDGE + 255) / 256, 256, 0, stream>>>(logit, dst, mbuf, ex, den);
  k_scatter<<<NEDGE / 16, 160, 0, stream>>>(sat_out, w2, ex, den, dst, hagg);
  k_concentrate<<<NNODE / 16, 256, 0, stream>>>(hagg, hn, W_concentrate, b_concentrate,
                                                deg, hnew_out);
  k_aggre<<<NEDGE / 16, 256, 0, stream>>>(hnew_out, src, dst, w2, ow, bn_gamma, bn_beta,
                                          bn_mean, bn_var, W_aggre, wout_out);
  (void)in_sizes; (void)n_in; (void)out_size; (void)ws_size;
}